// PointPillarsScatter_14937896255768
// MI455X (gfx1250) — compile-verified
//
#include <hip/hip_runtime.h>

#ifndef __has_builtin
#define __has_builtin(x) 0
#endif

// Problem constants (match reference.py)
#define NYv   496
#define NXv   432
#define CCH   64
#define PT    48000
#define BATCH 4
#define OUT_ELEMS (BATCH * CCH * NYv * NXv)   // 54,853,632 floats (~219 MB)

// The gfx1250 async builtins take pointers to 4xi32 vectors:
//   void __builtin_amdgcn_global_store_async_from_lds_b128(
//        v4i addrspace(1)* gdst, v4i addrspace(3)* lsrc, imm i32 off, imm i32 cpol)
typedef int v4i __attribute__((vector_size(4 * sizeof(int))));
typedef __attribute__((address_space(1))) v4i gv4i;   // global 16B vector
typedef __attribute__((address_space(3))) v4i lv4i;   // LDS 16B vector

// CDNA5 async LDS->global store path (ASYNCcnt). Guarded so the file always
// compiles; if the builtins are missing we fall back to plain b128 stores.
#if defined(__gfx1250__) && \
    __has_builtin(__builtin_amdgcn_global_store_async_from_lds_b128) && \
    __has_builtin(__builtin_amdgcn_s_wait_asynccnt)
#define HAVE_ASYNC_LDS 1
#else
#define HAVE_ASYNC_LDS 0
#endif

// ---------------------------------------------------------------------------
// Kernel 1: zero the [B, C, NY, NX] canvas with 16B stores.
// Async path: zero a 4 KB LDS staging buffer once per block, then each lane
// issues GLOBAL_STORE_ASYNC_FROM_LDS_B128 (512 B per wave per instruction).
// HW stalls issue if ASYNCcnt would overflow, so no manual throttling needed;
// we drain with s_wait_asynccnt 0 before exit (S_ENDPGM also implies
// wait-idle, this is belt-and-braces).
// ---------------------------------------------------------------------------
__global__ __launch_bounds__(256) void pp_zero_canvas_kernel(float4* __restrict__ out4,
                                                             int n4) {
  const int tid    = blockIdx.x * blockDim.x + threadIdx.x;
  const int stride = gridDim.x * blockDim.x;
#if HAVE_ASYNC_LDS
  __shared__ __align__(16) float4 zbuf[256];
  zbuf[threadIdx.x] = make_float4(0.f, 0.f, 0.f, 0.f);
  __syncthreads();  // LDS zeros visible before the async engine reads them
  lv4i* lsrc = (lv4i*)&zbuf[threadIdx.x];  // distinct 16B per lane
  for (int i = tid; i < n4; i += stride) {
    __builtin_amdgcn_global_store_async_from_lds_b128(
        (gv4i*)(out4 + i), lsrc, /*imm_offset=*/0, /*cpol=*/0);
  }
  __builtin_amdgcn_s_wait_asynccnt(0);
#else
  const float4 z = make_float4(0.f, 0.f, 0.f, 0.f);
  for (int i = tid; i < n4; i += stride) out4[i] = z;
#endif
}

// ---------------------------------------------------------------------------
// Kernel 2: scatter. Thread t handles point p = t>>6, channel c = t&63.
//   out[((b*64 + c)*496 + y)*432 + x] = vf[p*64 + c]
// Feature reads are fully coalesced; coords are a wave-uniform broadcast
// (each wave covers 32 channels of exactly one point). Stores are inherently
// scattered b32 — absorbed by the 192 MB L2 since the canvas was just zeroed.
// Duplicate flat indices resolve "one wins", same as jax scatter-set.
// ---------------------------------------------------------------------------
__global__ __launch_bounds__(256) void pp_scatter_kernel(const float* __restrict__ vf,
                                                         const int*   __restrict__ coords,
                                                         float*       __restrict__ out) {
  const int t = blockIdx.x * blockDim.x + threadIdx.x;
  if (t >= PT * CCH) return;
  const int p = t >> 6;           // point index
  const int c = t & (CCH - 1);    // channel
  const int b = coords[4 * p + 0];
  const int y = coords[4 * p + 2];
  const int x = coords[4 * p + 3];
  const float v = vf[t];          // vf[p*64 + c] — coalesced
  const int oidx = ((b * CCH + c) * NYv + y) * NXv + x;  // < 2^31, 32-bit ok
  out[oidx] = v;
}

// ---------------------------------------------------------------------------
// Launch: inputs are (voxel_features f32[48000*64], coords i32[48000*4],
// batch_size i32[1] — fixed at 4 per the reference constants).
// ---------------------------------------------------------------------------
extern "C" void kernel_launch(void* const* d_in, const int* in_sizes, int n_in,
                              void* d_out, int out_size, void* d_ws, size_t ws_size,
                              hipStream_t stream) {
  const float* vf     = (const float*)d_in[0];
  const int*   coords = (const int*)d_in[1];
  float*       out    = (float*)d_out;

  const int n4 = OUT_ELEMS / 4;  // 13,713,408 float4 stores
  hipLaunchKernelGGL(pp_zero_canvas_kernel, dim3(4096), dim3(256), 0, stream,
                     (float4*)out, n4);

  const int total  = PT * CCH;                 // 3,072,000
  const int sgrid  = (total + 255) / 256;      // 12,000 blocks
  hipLaunchKernelGGL(pp_scatter_kernel, dim3(sgrid), dim3(256), 0, stream,
                     vf, coords, out);
}